// SAGE_51788715655557
// MI455X (gfx1250) — compile-verified
//
#include <hip/hip_runtime.h>

typedef __attribute__((ext_vector_type(2))) float v2f;
typedef __attribute__((ext_vector_type(8))) float v8f;

#define FEAT 128

// ---------------------------------------------------------------- utilities
__global__ void zero_f32(float* __restrict__ p, size_t n) {
  size_t i = (size_t)blockIdx.x * blockDim.x + threadIdx.x;
  size_t stride = (size_t)gridDim.x * blockDim.x;
  for (; i < n; i += stride) p[i] = 0.0f;
}

__global__ void degree_kernel(const int* __restrict__ dst, float* __restrict__ deg, int E) {
  int i = blockIdx.x * blockDim.x + threadIdx.x;
  int stride = gridDim.x * blockDim.x;
  for (; i < E; i += stride) atomicAdd(&deg[dst[i]], 1.0f);
}

__global__ void invdeg_kernel(float* __restrict__ deg, int n) {
  int i = blockIdx.x * blockDim.x + threadIdx.x;
  int stride = gridDim.x * blockDim.x;
  for (; i < n; i += stride) deg[i] = 1.0f / fmaxf(deg[i], 1.0f);
}

// --------------------------------------------------- mean-aggregation scatter
// One wave (32 lanes) per edge; lane handles 4 consecutive features (float4).
// Value is pre-scaled by inv_deg[dst] so agg ends up as the neighbor mean.
__global__ void scatter_mean_kernel(const float* __restrict__ h,
                                    const int* __restrict__ src,
                                    const int* __restrict__ dst,
                                    const float* __restrict__ invdeg,
                                    float* __restrict__ agg, int E) {
  const int lane = threadIdx.x & 31;
  const int wave = (blockIdx.x * blockDim.x + threadIdx.x) >> 5;
  const int nwaves = (gridDim.x * blockDim.x) >> 5;
  for (int e = wave; e < E; e += nwaves) {
    const int s = src[e];
    const int t = dst[e];
    const float w = invdeg[t];
    const float4 v = *(const float4*)(h + (size_t)s * FEAT + lane * 4);
    float* ap = agg + (size_t)t * FEAT + lane * 4;
    atomicAdd(ap + 0, v.x * w);
    atomicAdd(ap + 1, v.y * w);
    atomicAdd(ap + 2, v.z * w);
    atomicAdd(ap + 3, v.w * w);
  }
}

// ------------------------------------------------------------- fused SAGE GEMM
// out = relu?( agg @ Wl + b + h @ Wr )
// 8 waves per block; wave j owns output columns [16j, 16j+16).
// Wave preloads B fragments of Wl and Wr for all K=128 (32 v2f each) into
// VGPRs, then grid-strides over 16-row M tiles staged through LDS.
__launch_bounds__(256)
__global__ void sage_gemm_kernel(const float* __restrict__ Aagg,
                                 const float* Ah,          // may alias out (tilewise safe)
                                 const float* __restrict__ Wl,
                                 const float* __restrict__ bias,
                                 const float* __restrict__ Wr,
                                 float* out,
                                 int n_nodes, int do_relu) {
  __shared__ float sAgg[16][132];   // +4 pad: 16B-aligned rows, 64-bank conflict free
  __shared__ float sH[16][132];

  const int lane = threadIdx.x & 31;
  const int wave = threadIdx.x >> 5;    // 0..7
  const int c    = lane & 15;           // col within 16-tile (== row m for A frag)
  const int half = lane >> 4;           // selects K pair (A/B) / row half (C)
  const int n0   = wave * 16;

  // ---- preload B fragments: B[k][n], lane = half*16 + n, vgpr = k & 1
  v2f bl[32], br[32];
  {
    const float* wl = Wl + n0 + c;
    const float* wr = Wr + n0 + c;
#pragma unroll
    for (int s = 0; s < 32; ++s) {
      const int k = 4 * s + half * 2;
      v2f t0; t0.x = wl[(size_t)k * FEAT]; t0.y = wl[(size_t)(k + 1) * FEAT];
      bl[s] = t0;
      v2f t1; t1.x = wr[(size_t)k * FEAT]; t1.y = wr[(size_t)(k + 1) * FEAT];
      br[s] = t1;
    }
  }
  const float bv = bias[n0 + c];
  const int ntiles = (n_nodes + 15) >> 4;

  for (int tile = blockIdx.x; tile < ntiles; tile += gridDim.x) {
    const int row0 = tile << 4;
    __syncthreads();  // previous iteration's LDS readers done
    // cooperative stage of 16x128 agg + self tiles (512 float4 each)
    for (int i = threadIdx.x; i < 512; i += 256) {
      const int r  = i >> 5;
      const int c4 = (i & 31) << 2;
      float4 va = make_float4(0.f, 0.f, 0.f, 0.f), vh = va;
      if (row0 + r < n_nodes) {
        const size_t g = (size_t)(row0 + r) * FEAT + c4;
        va = *(const float4*)(Aagg + g);
        vh = *(const float4*)(Ah + g);
      }
      *(float4*)&sAgg[r][c4] = va;
      *(float4*)&sH[r][c4]   = vh;
    }
    __syncthreads();

    v8f acc = {};  // 16x16 f32 accumulator (8 VGPRs)
#pragma unroll
    for (int s = 0; s < 32; ++s) {
      const int k = 4 * s + half * 2;
      // A frag: A[m][k..k+1], lane = half*16 + m  ->  m == c here
      v2f a0 = *(const v2f*)&sAgg[c][k];
      acc = __builtin_amdgcn_wmma_f32_16x16x4_f32(false, a0, false, bl[s],
                                                  (short)0, acc, false, false);
      v2f a1 = *(const v2f*)&sH[c][k];
      acc = __builtin_amdgcn_wmma_f32_16x16x4_f32(false, a1, false, br[s],
                                                  (short)0, acc, false, false);
    }

    // C layout: vgpr r holds row (r + 8*half), col = lane&15
#pragma unroll
    for (int r = 0; r < 8; ++r) {
      const int m = row0 + r + half * 8;
      if (m < n_nodes) {
        float v = acc[r] + bv;
        if (do_relu) v = fmaxf(v, 0.0f);
        out[(size_t)m * FEAT + n0 + c] = v;
      }
    }
  }
}

// -------------------------------------------------------------------- launch
extern "C" void kernel_launch(void* const* d_in, const int* in_sizes, int n_in,
                              void* d_out, int out_size, void* d_ws, size_t ws_size,
                              hipStream_t stream) {
  const float* x  = (const float*)d_in[0];
  const int*   ei = (const int*)d_in[1];     // edge_index, int32, shape [2, E]
  const int N = in_sizes[0] / FEAT;
  const int E = in_sizes[1] / 2;
  const int* src = ei;
  const int* dst = ei + E;

  const float* Wl[3] = {(const float*)d_in[2], (const float*)d_in[5], (const float*)d_in[8]};
  const float* bb[3] = {(const float*)d_in[3], (const float*)d_in[6], (const float*)d_in[9]};
  const float* Wr[3] = {(const float*)d_in[4], (const float*)d_in[7], (const float*)d_in[10]};

  float* deg = (float*)d_ws;                 // N floats (becomes inv_deg)
  float* agg = deg + N;                      // N*FEAT floats
  float* h   = agg + (size_t)N * FEAT;       // N*FEAT floats
  float* outf = (float*)d_out;

  // degree -> inverse degree (recomputed every call: deterministic)
  zero_f32<<<512, 256, 0, stream>>>(deg, (size_t)N);
  degree_kernel<<<2048, 256, 0, stream>>>(dst, deg, E);
  invdeg_kernel<<<512, 256, 0, stream>>>(deg, N);

  const float* hin = x;
  float* houts[3] = {h, h, outf};
  for (int l = 0; l < 3; ++l) {
    zero_f32<<<4096, 256, 0, stream>>>(agg, (size_t)N * FEAT);
    scatter_mean_kernel<<<8192, 256, 0, stream>>>(hin, src, dst, deg, agg, E);
    sage_gemm_kernel<<<1024, 256, 0, stream>>>(agg, hin, Wl[l], bb[l], Wr[l],
                                               houts[l], N, (l < 2) ? 1 : 0);
    hin = houts[l];
  }
}